// Petri_GCN_74921409511934
// MI455X (gfx1250) — compile-verified
//
#include <hip/hip_runtime.h>
#include <hip/hip_bf16.h>

typedef __attribute__((ext_vector_type(2))) float v2f;
typedef __attribute__((ext_vector_type(8))) float v8f;

#define HIDC 64

// ---------------------------------------------------------------------------
// Degree accumulation: deg[dst] += w  (one thread per edge)
// ---------------------------------------------------------------------------
__global__ __launch_bounds__(256)
void gcn_deg_kernel(const int* __restrict__ dst, const float* __restrict__ w,
                    float* __restrict__ deg, int E) {
    int e = blockIdx.x * blockDim.x + threadIdx.x;
    if (e < E) atomicAdd(&deg[dst[e]], w[e]);
}

// ---------------------------------------------------------------------------
// Per-node prep: invd[i] = 1/(deg+1), dinv[i] = rsqrt(deg+1). deg buffer is
// overwritten with invd.
// ---------------------------------------------------------------------------
__global__ __launch_bounds__(256)
void gcn_prep_kernel(float* __restrict__ deg_inv, float* __restrict__ dinv, int n) {
    int i = blockIdx.x * blockDim.x + threadIdx.x;
    if (i < n) {
        float dg = deg_inv[i] + 1.0f;
        dinv[i] = rsqrtf(dg);
        deg_inv[i] = 1.0f / dg;
    }
}

// ---------------------------------------------------------------------------
// Y[N,64] = X[N,64] @ W[64,64]  via V_WMMA_F32_16X16X4_F32.
// One wave computes a 16-row x 64-col tile: 4 v8f accumulators, K=64 in 16
// steps of 4. A-layout (ISA 7.12.2, 32-bit A 16x4): lane l (0..15) holds
// row M=l, K = kb+{0,1}; lane l+16 holds K = kb+{2,3}. B/C/D: row striped
// across lanes within a VGPR; C/D VGPR j -> M=j (lanes 0-15), M=j+8 (16-31).
// ---------------------------------------------------------------------------
__global__ __launch_bounds__(256)
void gcn_gemm64_wmma(const float* __restrict__ X, const float* __restrict__ W,
                     float* __restrict__ Y, int n) {
    const int lane = threadIdx.x & 31;
    const int wave = blockIdx.x * (blockDim.x >> 5) + (threadIdx.x >> 5);
    const int nTiles = (n + 15) >> 4;
    if (wave >= nTiles) return;              // wave-uniform: EXEC stays all-1s

    const int row0 = wave << 4;
    const int lm = lane & 15;
    const int kh = (lane >> 4) << 1;         // 0 for lanes 0-15, 2 for 16-31

    int arow = row0 + lm;
    if (arow >= n) arow = n - 1;             // clamp (N=100000 is 16-aligned anyway)
    const float* __restrict__ xrow = X + (size_t)arow * HIDC;

    v8f acc0 = {}, acc1 = {}, acc2 = {}, acc3 = {};

#pragma unroll
    for (int kb = 0; kb < HIDC; kb += 4) {
        v2f a;
        a.x = xrow[kb + kh];
        a.y = xrow[kb + kh + 1];

        const float* __restrict__ w0 = W + (size_t)(kb + kh) * HIDC;   // K row kb+kh
        const float* __restrict__ w1 = w0 + HIDC;                      // K row kb+kh+1
        v2f b0, b1, b2, b3;
        b0.x = w0[lm     ]; b0.y = w1[lm     ];
        b1.x = w0[lm + 16]; b1.y = w1[lm + 16];
        b2.x = w0[lm + 32]; b2.y = w1[lm + 32];
        b3.x = w0[lm + 48]; b3.y = w1[lm + 48];

        acc0 = __builtin_amdgcn_wmma_f32_16x16x4_f32(false, a, false, b0, (short)0, acc0, false, false);
        acc1 = __builtin_amdgcn_wmma_f32_16x16x4_f32(false, a, false, b1, (short)0, acc1, false, false);
        acc2 = __builtin_amdgcn_wmma_f32_16x16x4_f32(false, a, false, b2, (short)0, acc2, false, false);
        acc3 = __builtin_amdgcn_wmma_f32_16x16x4_f32(false, a, false, b3, (short)0, acc3, false, false);
    }

    const int rbase = (lane >> 4) << 3;      // 0 or 8
#pragma unroll
    for (int j = 0; j < 8; j++) {
        int r = row0 + rbase + j;
        if (r < n) {
            float* __restrict__ yr = Y + (size_t)r * HIDC;
            yr[lm     ] = acc0[j];
            yr[lm + 16] = acc1[j];
            yr[lm + 32] = acc2[j];
            yr[lm + 48] = acc3[j];
        }
    }
}

// ---------------------------------------------------------------------------
// Edge scatter: one wave per edge, lane covers features [2*lane, 2*lane+1].
// agg[dst] += xw[src] * (dinv[src]*w*dinv[dst]). Coalesced 256B per edge.
// ---------------------------------------------------------------------------
__global__ __launch_bounds__(256)
void gcn_scatter_kernel(const int* __restrict__ src, const int* __restrict__ dst,
                        const float* __restrict__ w, const float* __restrict__ dinv,
                        const float* __restrict__ xw, float* __restrict__ agg, int E) {
    const int e = blockIdx.x * (blockDim.x >> 5) + (threadIdx.x >> 5);
    const int lane = threadIdx.x & 31;
    if (e >= E) return;
    const int s = src[e];
    const int d = dst[e];
    const float nrm = dinv[s] * w[e] * dinv[d];
    const float2 v = *(const float2*)(xw + (size_t)s * HIDC + lane * 2);
    float* __restrict__ ap = agg + (size_t)d * HIDC + lane * 2;
    atomicAdd(ap,     v.x * nrm);
    atomicAdd(ap + 1, v.y * nrm);
}

// ---------------------------------------------------------------------------
// Combine: out = agg + xw * invd[node] + bias, optional ReLU. float4 per thread.
// ---------------------------------------------------------------------------
__global__ __launch_bounds__(256)
void gcn_combine_kernel(float* __restrict__ agg, const float* __restrict__ xw,
                        const float* __restrict__ invd, const float* __restrict__ bias,
                        int n, int relu) {
    int t = blockIdx.x * blockDim.x + threadIdx.x;      // over n*16 float4s
    if (t >= n * 16) return;
    const int node = t >> 4;
    const int c = (t & 15) << 2;
    const float inv = invd[node];
    const size_t off = (size_t)node * HIDC + c;
    float4 a = *(const float4*)(agg + off);
    float4 x = *(const float4*)(xw + off);
    float4 b = *(const float4*)(bias + c);
    a.x += x.x * inv + b.x;
    a.y += x.y * inv + b.y;
    a.z += x.z * inv + b.z;
    a.w += x.w * inv + b.w;
    if (relu) {
        a.x = fmaxf(a.x, 0.f); a.y = fmaxf(a.y, 0.f);
        a.z = fmaxf(a.z, 0.f); a.w = fmaxf(a.w, 0.f);
    }
    *(float4*)(agg + off) = a;
}

// ---------------------------------------------------------------------------
// Readout MLP + pooling: one wave per node. Lane j computes hidden unit j:
// relu(sum_c h[c]*Wr0[c,j] + br0[j]) * Wr1[j]; shuffle-reduce; lane0 pools.
// ---------------------------------------------------------------------------
__global__ __launch_bounds__(256)
void gcn_readout_kernel(const float* __restrict__ h, const float* __restrict__ Wr0,
                        const float* __restrict__ br0, const float* __restrict__ Wr1,
                        const float* __restrict__ br1, const int* __restrict__ batch,
                        float* __restrict__ sums, float* __restrict__ cnt, int n) {
    const int node = blockIdx.x * (blockDim.x >> 5) + (threadIdx.x >> 5);
    const int lane = threadIdx.x & 31;
    if (node >= n) return;
    const float* __restrict__ hp = h + (size_t)node * HIDC;
    float acc = br0[lane];
#pragma unroll 8
    for (int c = 0; c < HIDC; c++)
        acc += hp[c] * Wr0[c * 32 + lane];      // Wr0 is [64,32]
    acc = fmaxf(acc, 0.f);
    float r = acc * Wr1[lane];                  // Wr1 is [32,1]
#pragma unroll
    for (int off = 16; off > 0; off >>= 1)
        r += __shfl_xor(r, off, 32);
    if (lane == 0) {
        const int g = batch[node];
        atomicAdd(&sums[g], r + br1[0]);
        atomicAdd(&cnt[g], 1.0f);
    }
}

__global__ __launch_bounds__(256)
void gcn_finalize_kernel(const float* __restrict__ sums, const float* __restrict__ cnt,
                         float* __restrict__ out, int g) {
    int i = blockIdx.x * blockDim.x + threadIdx.x;
    if (i < g) out[i] = sums[i] / fmaxf(cnt[i], 1.0f);
}

// ---------------------------------------------------------------------------
extern "C" void kernel_launch(void* const* d_in, const int* in_sizes, int n_in,
                              void* d_out, int out_size, void* d_ws, size_t ws_size,
                              hipStream_t stream) {
    const float* x     = (const float*)d_in[0];
    const int*   ei    = (const int*)  d_in[1];
    const float* ew    = (const float*)d_in[2];
    const int*   batch = (const int*)  d_in[3];
    const float* W0 = (const float*)d_in[4];  const float* b0 = (const float*)d_in[5];
    const float* W1 = (const float*)d_in[6];  const float* b1 = (const float*)d_in[7];
    const float* W2 = (const float*)d_in[8];  const float* b2 = (const float*)d_in[9];
    const float* Wr0 = (const float*)d_in[10]; const float* br0 = (const float*)d_in[11];
    const float* Wr1 = (const float*)d_in[12]; const float* br1 = (const float*)d_in[13];
    float* out = (float*)d_out;

    const int N = in_sizes[0] / HIDC;     // 100000
    const int E = in_sizes[2];            // 1600000
    const int G = out_size;               // 256

    const int* src = ei;
    const int* dst = ei + E;

    // workspace layout (floats)
    float* bufA = (float*)d_ws;                    // xw scratch          N*64
    float* bufB = bufA + (size_t)N * HIDC;         // agg / layer out     N*64
    float* bufC = bufB + (size_t)N * HIDC;         // agg / layer out     N*64
    float* invd = bufC + (size_t)N * HIDC;         // deg -> 1/deg        N
    float* dinv = invd + N;                        // rsqrt(deg)          N
    float* sums = dinv + N;                        // G
    float* cnt  = sums + G;                        // G

    const int TPB = 256;
    const int wavesPerBlk = TPB / 32;
    const size_t featBytes = (size_t)N * HIDC * sizeof(float);

    const int gemmBlocks    = ((N + 15) / 16 + wavesPerBlk - 1) / wavesPerBlk;
    const int scatterBlocks = (E + wavesPerBlk - 1) / wavesPerBlk;
    const int combineBlocks = (N * 16 + TPB - 1) / TPB;
    const int nodeBlocks    = (N + TPB - 1) / TPB;
    const int edgeBlocks    = (E + TPB - 1) / TPB;
    const int rdBlocks      = (N + wavesPerBlk - 1) / wavesPerBlk;

    // degrees
    hipMemsetAsync(invd, 0, (size_t)N * sizeof(float), stream);
    gcn_deg_kernel<<<edgeBlocks, TPB, 0, stream>>>(dst, ew, invd, E);
    gcn_prep_kernel<<<nodeBlocks, TPB, 0, stream>>>(invd, dinv, N);

    // layer 0: in = x, out -> bufB (ReLU)
    gcn_gemm64_wmma<<<gemmBlocks, TPB, 0, stream>>>(x, W0, bufA, N);
    hipMemsetAsync(bufB, 0, featBytes, stream);
    gcn_scatter_kernel<<<scatterBlocks, TPB, 0, stream>>>(src, dst, ew, dinv, bufA, bufB, E);
    gcn_combine_kernel<<<combineBlocks, TPB, 0, stream>>>(bufB, bufA, invd, b0, N, 1);

    // layer 1: in = bufB, out -> bufC (ReLU)
    gcn_gemm64_wmma<<<gemmBlocks, TPB, 0, stream>>>(bufB, W1, bufA, N);
    hipMemsetAsync(bufC, 0, featBytes, stream);
    gcn_scatter_kernel<<<scatterBlocks, TPB, 0, stream>>>(src, dst, ew, dinv, bufA, bufC, E);
    gcn_combine_kernel<<<combineBlocks, TPB, 0, stream>>>(bufC, bufA, invd, b1, N, 1);

    // layer 2: in = bufC, out -> bufB (no ReLU)
    gcn_gemm64_wmma<<<gemmBlocks, TPB, 0, stream>>>(bufC, W2, bufA, N);
    hipMemsetAsync(bufB, 0, featBytes, stream);
    gcn_scatter_kernel<<<scatterBlocks, TPB, 0, stream>>>(src, dst, ew, dinv, bufA, bufB, E);
    gcn_combine_kernel<<<combineBlocks, TPB, 0, stream>>>(bufB, bufA, invd, b2, N, 0);

    // readout + mean pool
    hipMemsetAsync(sums, 0, (size_t)G * sizeof(float), stream);
    hipMemsetAsync(cnt,  0, (size_t)G * sizeof(float), stream);
    gcn_readout_kernel<<<rdBlocks, TPB, 0, stream>>>(bufB, Wr0, br0, Wr1, br1, batch, sums, cnt, N);
    gcn_finalize_kernel<<<(G + TPB - 1) / TPB, TPB, 0, stream>>>(sums, cnt, out, G);
}